// CSIW_77687368450668
// MI455X (gfx1250) — compile-verified
//
#include <hip/hip_runtime.h>
#include <hip/hip_bf16.h>

typedef __attribute__((ext_vector_type(2))) float v2f;
typedef __attribute__((ext_vector_type(4))) float v4f;
typedef __attribute__((ext_vector_type(8))) float v8f;

#define C_DIM 640
#define M_DIM 25
#define QBATCH 4096
#define WS_ROWS 25

// ---------------------------------------------------------------------------
// Kernel 1: mean over trailing m=25 axis.  x: (R, 640, 25) -> out: (R, 640)
// Each thread owns one (r,c) pair: 100 contiguous bytes; waves cover
// contiguous 3.2 KB stretches so cache lines are fully used.
// ---------------------------------------------------------------------------
__global__ __launch_bounds__(256) void mean25_kernel(const float* __restrict__ x,
                                                     float* __restrict__ out,
                                                     int total) {
  int idx = blockIdx.x * 256 + threadIdx.x;
  if (idx >= total) return;
  const float* p = x + (size_t)idx * M_DIM;
  float s = 0.f;
#pragma unroll
  for (int m = 0; m < M_DIM; ++m) s += p[m];
  out[idx] = s * (1.0f / 25.0f);
}

// ---------------------------------------------------------------------------
// Kernel 2: WMMA f32 GEMM + BN + (1 + tanh) epilogue.
//   Out[m,n] = 1 + tanh( (sum_k A[m,k]*W[n,k] - rm[n]) * g[n]*rsqrt(rv[n]+eps) + b[n] )
// One wave per 16x16 output tile; K advances 4 per v_wmma_f32_16x16x4_f32.
// f32 A fragment (16x4): lanes 0-15 hold K=0..1, lanes 16-31 hold K=2..3 of the
// same rows -> one contiguous 8B load per lane.  B fragment (4x16) mirrors it
// on W's rows (W is N x K row-major, so both operands are K-contiguous).
// ---------------------------------------------------------------------------
__global__ __launch_bounds__(256) void gemm_bn_tanh_kernel(
    const float* __restrict__ A, const float* __restrict__ W,
    const float* __restrict__ g, const float* __restrict__ bb,
    const float* __restrict__ rm, const float* __restrict__ rv,
    float* __restrict__ out, int M, int tilesM) {
  const int TN = C_DIM / 16;               // 40 tiles along N
  int wave = threadIdx.x >> 5;
  int lane = threadIdx.x & 31;
  int tile = blockIdx.x * 8 + wave;
  if (tile >= tilesM * TN) return;         // wave-uniform: EXEC stays all-ones
  int tm = tile / TN;
  int tn = tile % TN;
  int half = lane >> 4;                    // 0: K=0..1 / rows 0..7, 1: K=2..3 / rows 8..15
  int l16  = lane & 15;

  int arow  = tm * 16 + l16;
  int arowc = (arow < M) ? arow : 0;       // clamp padded rows (M=25 case)
  const float* Ap = A + (size_t)arowc * C_DIM + 2 * half;
  const float* Wp = W + (size_t)(tn * 16 + l16) * C_DIM + 2 * half;

  v8f acc = {};
#pragma unroll 4
  for (int k = 0; k < C_DIM; k += 4) {
    v2f av = *(const v2f*)(Ap + k);
    v2f bv = *(const v2f*)(Wp + k);
    acc = __builtin_amdgcn_wmma_f32_16x16x4_f32(
        /*neg_a=*/false, av, /*neg_b=*/false, bv,
        /*c_mod=*/(short)0, acc, /*reuse_a=*/false, /*reuse_b=*/false);
  }

  int n = tn * 16 + l16;                   // column for all 8 result VGPRs
  float sc = g[n] * rsqrtf(rv[n] + 1e-5f);
  float sh = bb[n] - rm[n] * sc;
#pragma unroll
  for (int r = 0; r < 8; ++r) {
    int m = tm * 16 + r + 8 * half;        // C/D layout: VGPR r -> row r (+8 for hi lanes)
    if (m < M) out[(size_t)m * C_DIM + n] = 1.0f + tanhf(acc[r] * sc + sh);
  }
}

// ---------------------------------------------------------------------------
// Kernel 3: per-row gating + SI block + negated squared distance.
//   scaled[c,m] = x[r,c,m] * gate[r,c]
//   pooled: max/mean over c per spatial m (5x5); 2->1ch 3x3 SAME conv (+cb),
//   sigmoid -> sig[25];  negd[r,c] = -sum_m (scaled[c,m] - sig[m])^2
// One workgroup (256 thr) per row; row staged in 64 KB LDS.
// ---------------------------------------------------------------------------
__global__ __launch_bounds__(256) void si_dist_kernel(
    const float* __restrict__ x, const float* __restrict__ gate,
    const float* __restrict__ cw, const float* __restrict__ cb,
    float* __restrict__ negd) {
  __shared__ float sx[C_DIM * M_DIM];      // 64000 B
  __shared__ float pooled[2 * M_DIM];
  __shared__ float sig[M_DIM];

  int r = blockIdx.x;
  const float* xr = x + (size_t)r * C_DIM * M_DIM;
  const float* gr = gate + (size_t)r * C_DIM;

  for (int i = threadIdx.x; i < C_DIM * M_DIM; i += 256) {
    int c = i / M_DIM;
    sx[i] = xr[i] * gr[c];
  }
  __syncthreads();

  if (threadIdx.x < M_DIM) {               // channel max / mean per spatial pos
    int m = threadIdx.x;
    float mx = -__builtin_inff(), sm = 0.f;
    for (int c = 0; c < C_DIM; ++c) {
      float v = sx[c * M_DIM + m];
      mx = fmaxf(mx, v);
      sm += v;
    }
    pooled[m] = mx;
    pooled[M_DIM + m] = sm * (1.0f / 640.0f);
  }
  __syncthreads();

  if (threadIdx.x < M_DIM) {               // 3x3 SAME cross-correlation + sigmoid
    int i = threadIdx.x / 5, j = threadIdx.x % 5;
    float acc = cb[0];
#pragma unroll
    for (int ci = 0; ci < 2; ++ci)
#pragma unroll
      for (int di = -1; di <= 1; ++di)
#pragma unroll
        for (int dj = -1; dj <= 1; ++dj) {
          int ii = i + di, jj = j + dj;
          if (ii >= 0 && ii < 5 && jj >= 0 && jj < 5)
            acc += cw[ci * 9 + (di + 1) * 3 + (dj + 1)] * pooled[ci * M_DIM + ii * 5 + jj];
        }
    sig[threadIdx.x] = 1.0f / (1.0f + __expf(-acc));
  }
  __syncthreads();

  for (int c = threadIdx.x; c < C_DIM; c += 256) {
    float d = 0.f;
#pragma unroll
    for (int m = 0; m < M_DIM; ++m) {
      float t = sx[c * M_DIM + m] - sig[m];
      d += t * t;
    }
    negd[(size_t)r * C_DIM + c] = -d;      // store negated: feeds s_block(-dist)
  }
}

// ---------------------------------------------------------------------------
// Kernel 4: out[i,b,c] = 0.5*wp[i,c] + 0.5*wq[b,c]   (25 x 4096 x 640)
// ext-vector float4 lanes; non-temporal stores for the 262 MB streaming output
// keep wq (10 MB, reused 25x) resident in L2.
// ---------------------------------------------------------------------------
__global__ __launch_bounds__(256) void bcast_add_kernel(
    const v4f* __restrict__ wp, const v4f* __restrict__ wq,
    v4f* __restrict__ out, long total4) {
  long idx = (long)blockIdx.x * 256 + threadIdx.x;
  if (idx >= total4) return;
  int  c4 = (int)(idx % 160);              // 640/4
  long rb = idx / 160;
  int  b  = (int)(rb & (QBATCH - 1));
  int  i  = (int)(rb >> 12);
  v4f a = wp[i * 160 + c4];
  v4f q = wq[(size_t)b * 160 + c4];
  v4f o = 0.5f * (a + q);
  __builtin_nontemporal_store(o, &out[idx]);
}

// ---------------------------------------------------------------------------
static void launch_gemm(const float* A, const float* W, const float* g,
                        const float* b, const float* rm, const float* rv,
                        float* out, int M, hipStream_t stream) {
  int tilesM = (M + 15) / 16;
  int tiles  = tilesM * (C_DIM / 16);
  int blocks = (tiles + 7) / 8;            // 8 waves (tiles) per 256-thread block
  gemm_bn_tanh_kernel<<<blocks, 256, 0, stream>>>(A, W, g, b, rm, rv, out, M, tilesM);
}

extern "C" void kernel_launch(void* const* d_in, const int* in_sizes, int n_in,
                              void* d_out, int out_size, void* d_ws, size_t ws_size,
                              hipStream_t stream) {
  const float* spt   = (const float*)d_in[0];   // (5,5,640,25) == (25,640,25)
  const float* qry   = (const float*)d_in[1];   // (4096,1,640,25)
  const float* W_p   = (const float*)d_in[2];
  const float* g_p   = (const float*)d_in[3];
  const float* b_p   = (const float*)d_in[4];
  const float* rm_p  = (const float*)d_in[5];
  const float* rv_p  = (const float*)d_in[6];
  const float* W_q   = (const float*)d_in[7];
  const float* g_q   = (const float*)d_in[8];
  const float* b_q   = (const float*)d_in[9];
  const float* rm_q  = (const float*)d_in[10];
  const float* rv_q  = (const float*)d_in[11];
  const float* W_prt = (const float*)d_in[12];
  const float* g_prt = (const float*)d_in[13];
  const float* b_prt = (const float*)d_in[14];
  const float* rm_prt= (const float*)d_in[15];
  const float* rv_prt= (const float*)d_in[16];
  const float* W_qs  = (const float*)d_in[17];
  const float* g_qs  = (const float*)d_in[18];
  const float* b_qs  = (const float*)d_in[19];
  const float* rm_qs = (const float*)d_in[20];
  const float* rv_qs = (const float*)d_in[21];
  const float* cw_p  = (const float*)d_in[22];
  const float* cb_p  = (const float*)d_in[23];
  const float* cw_q  = (const float*)d_in[24];
  const float* cb_q  = (const float*)d_in[25];

  // Workspace layout (floats): ~42.3 MB total
  float* ws  = (float*)d_ws;
  const size_t QC = (size_t)QBATCH * C_DIM;   // 2,621,440
  const size_t SC = (size_t)WS_ROWS * C_DIM;  // 16,000
  float* xq  = ws;            // mean(qry)        QC
  float* qg  = xq  + QC;      // gate(qry)        QC
  float* nd2 = qg  + QC;      // -dist2           QC
  float* wq  = nd2 + QC;      // w_qry            QC
  float* xs  = wq  + QC;      // mean(spt)        SC
  float* sg  = xs  + SC;      // gate(spt)        SC
  float* nd1 = sg  + SC;      // -dist1           SC
  float* wp  = nd1 + SC;      // w_prt            SC

  // 1) means over m
  mean25_kernel<<<(int)((QC + 255) / 256), 256, 0, stream>>>(qry, xq, (int)QC);
  mean25_kernel<<<(int)((SC + 255) / 256), 256, 0, stream>>>(spt, xs, (int)SC);

  // 2) gating s_blocks (WMMA GEMM + BN + 1+tanh)
  launch_gemm(xq, W_q, g_q, b_q, rm_q, rv_q, qg, QBATCH, stream);
  launch_gemm(xs, W_p, g_p, b_p, rm_p, rv_p, sg, WS_ROWS, stream);

  // 3) scale + SI block + negated distances
  si_dist_kernel<<<QBATCH, 256, 0, stream>>>(qry, qg, cw_q, cb_q, nd2);
  si_dist_kernel<<<WS_ROWS, 256, 0, stream>>>(spt, sg, cw_p, cb_p, nd1);

  // 4) weight s_blocks on -dist
  launch_gemm(nd2, W_qs, g_qs, b_qs, rm_qs, rv_qs, wq, QBATCH, stream);
  launch_gemm(nd1, W_prt, g_prt, b_prt, rm_prt, rv_prt, wp, WS_ROWS, stream);

  // 5) broadcast combine -> (25, 4096, 640)
  long total4 = (long)WS_ROWS * QBATCH * (C_DIM / 4);  // 16,384,000
  bcast_add_kernel<<<(int)((total4 + 255) / 256), 256, 0, stream>>>(
      (const v4f*)wp, (const v4f*)wq, (v4f*)d_out, total4);
}